// CentroidLoss_87540023427571
// MI455X (gfx1250) — compile-verified
//
#include <hip/hip_runtime.h>

typedef __attribute__((ext_vector_type(2))) float v2f;
typedef __attribute__((ext_vector_type(8))) float v8f;

constexpr int BN = 8, H = 512, W = 512, HW = H * W, L = 64;
constexpr float INVN = 1.0f / (float)(BN * HW);    // mean over B*H*W
// exp(-d2/(2*sigma^2)) = exp2(d2 * EXPC), sigma=2 -> -0.125 * log2(e)
constexpr float EXPC = -0.125f * 1.4426950408889634f;

// Compile-time ds_swizzle (group-of-32 XOR mode): lane -> lane ^ XORMASK
template <int XORMASK>
__device__ __forceinline__ float swz_xor(float v) {
    constexpr int pat = (XORMASK << 10) | 0x1f;   // offset[14:10]=xor, [4:0]=and
    int r = __builtin_amdgcn_ds_swizzle(__builtin_bit_cast(int, v), pat);
    return __builtin_bit_cast(float, r);
}

// ---------------- Phase 0: zero accumulators + output ----------------
__global__ void centroid_init(int* wsInt, float* out) {
    int i = blockIdx.x * blockDim.x + threadIdx.x;
    if (i < BN * L * 3) wsInt[i] = 0;
    if (i == 0) out[0] = 0.0f;
}

// ---------------- Phase 1: per-(batch,label) coordinate sums ----------------
__global__ __launch_bounds__(256)
void centroid_accum(const int* __restrict__ mask, int* __restrict__ cnt,
                    int* __restrict__ sy, int* __restrict__ sx) {
    __shared__ int sc[L], syy[L], sxx[L];
    const int tid = threadIdx.x;
    const int b = blockIdx.x >> 6;      // 64 chunks per batch
    const int chunk = blockIdx.x & 63;  // 4096 pixels per chunk
    if (tid < L) { sc[tid] = 0; syy[tid] = 0; sxx[tid] = 0; }
    __syncthreads();
    const int base = chunk * 4096;
    #pragma unroll 4
    for (int it = 0; it < 16; ++it) {
        int p = base + it * 256 + tid;
        int lab = mask[b * HW + p] & (L - 1);
        atomicAdd(&sc[lab], 1);
        atomicAdd(&syy[lab], p >> 9);   // y
        atomicAdd(&sxx[lab], p & 511);  // x
    }
    __syncthreads();
    if (tid < L) {
        atomicAdd(&cnt[b * L + tid], sc[tid]);
        atomicAdd(&sy[b * L + tid], syy[tid]);
        atomicAdd(&sx[b * L + tid], sxx[tid]);
    }
}

// ---------------- Phase 2: centroid table as WMMA A-rows ----------------
// table[b*L + l] = { -2*cx, -2*cy, cx^2+cy^2, 0 }, invalid -> c2 = 1e30 (exp underflows to 0)
__global__ void centroid_table(const int* __restrict__ cnt, const int* __restrict__ sy,
                               const int* __restrict__ sx, float4* __restrict__ table) {
    int i = blockIdx.x * blockDim.x + threadIdx.x;
    if (i >= BN * L) return;
    int l = i & (L - 1);
    int c = cnt[i];
    float4 e;
    if (c > 0 && l > 0) {
        float cy = (float)(sy[i] / c);  // floor for nonnegative ints
        float cx = (float)(sx[i] / c);
        e = make_float4(-2.0f * cx, -2.0f * cy, cx * cx + cy * cy, 0.0f);
    } else {
        e = make_float4(0.0f, 0.0f, 1.0e30f, 0.0f);
    }
    table[i] = e;
}

// ---------------- Phase 3: WMMA distance-min + MSE reduction ----------------
// D(labels x pixels) = A(labels x 4) * B(4 x pixels); min over labels = per-lane
// min over 8 D-VGPRs + one lane^16 swizzle. Then d2 = mind + (x^2+y^2).
__global__ __launch_bounds__(256)
void centroid_main(const float* __restrict__ pred, const float4* __restrict__ table,
                   float* __restrict__ out) {
    const int lane = threadIdx.x & 31;
    const int wave = threadIdx.x >> 5;
    const int b = blockIdx.x >> 5;      // 32 chunks per batch
    const int chunk = blockIdx.x & 31;
    const int n = lane & 15;
    const bool lo = lane < 16;
    const float wmask = lo ? INVN : 0.0f;   // lanes>=16 duplicate pixels -> weight 0

    // A-matrix 16x4 f32 layout: v0 lanes0-15=K0 (M=lane), lanes16-31=K2; v1: K1 / K3.
    v2f A[4];
    #pragma unroll
    for (int t = 0; t < 4; ++t) {
        float4 e = table[b * L + t * 16 + n];
        A[t].x = lo ? e.x : e.z;   // K0 = -2cx   | K2 = cx^2+cy^2
        A[t].y = lo ? e.y : 0.0f;  // K1 = -2cy   | K3 = 0
    }

    v8f czero = {};
    float acc = 0.0f;

    const float k2term = lo ? 0.0f : 1.0f;  // B K2 row constant
    const int groupsPerChunk = (HW / 32) / 16;  // 512 groups of 16 pixels
    const float* predB = pred + b * HW;

    // Two groups per iteration for WMMA->VALU latency hiding.
    for (int g = wave; g < groupsPerChunk; g += 16) {
        #pragma unroll
        for (int u = 0; u < 2; ++u) {
            const int gg = g + u * 8;
            const int p0 = (chunk * groupsPerChunk + gg) * 16;  // same y for all 16 px
            const int y = p0 >> 9;
            const int x0 = p0 & 511;

            // B-matrix 4x16 f32: v0 lanes0-15=K0 row (N=lane), lanes16-31=K2; v1: K1/K3.
            v2f Bm;
            const float fx = (float)(x0 + n);
            const float fy = (float)y;
            Bm.x = lo ? fx : k2term;   // K0 = x | K2 = 1
            Bm.y = lo ? fy : 0.0f;     // K1 = y | K3 = 0

            v8f d0 = __builtin_amdgcn_wmma_f32_16x16x4_f32(
                false, A[0], false, Bm, (short)0, czero, false, false);
            v8f d1 = __builtin_amdgcn_wmma_f32_16x16x4_f32(
                false, A[1], false, Bm, (short)0, czero, false, false);
            v8f d2v = __builtin_amdgcn_wmma_f32_16x16x4_f32(
                false, A[2], false, Bm, (short)0, czero, false, false);
            v8f d3 = __builtin_amdgcn_wmma_f32_16x16x4_f32(
                false, A[3], false, Bm, (short)0, czero, false, false);

            float m0 = fminf(fminf(fminf(d0[0], d0[1]), fminf(d0[2], d0[3])),
                             fminf(fminf(d0[4], d0[5]), fminf(d0[6], d0[7])));
            float m1 = fminf(fminf(fminf(d1[0], d1[1]), fminf(d1[2], d1[3])),
                             fminf(fminf(d1[4], d1[5]), fminf(d1[6], d1[7])));
            float m2 = fminf(fminf(fminf(d2v[0], d2v[1]), fminf(d2v[2], d2v[3])),
                             fminf(fminf(d2v[4], d2v[5]), fminf(d2v[6], d2v[7])));
            float m3 = fminf(fminf(fminf(d3[0], d3[1]), fminf(d3[2], d3[3])),
                             fminf(fminf(d3[4], d3[5]), fminf(d3[6], d3[7])));
            float mind = fminf(fminf(m0, m1), fminf(m2, m3));

            // lanes n and n+16 hold complementary label halves for pixel n
            mind = fminf(mind, swz_xor<16>(mind));

            const float d2 = mind + (fx * fx + fy * fy);   // lanes>=16: garbage, masked
            const float heat = __builtin_amdgcn_exp2f(d2 * EXPC);
            const float pv = predB[p0 + n];                // hi lanes re-load same 16 addrs
            const float err = pv - heat;
            acc = fmaf(err * err, wmask, acc);
        }
    }

    // wave32 sum reduction via ds_swizzle xor steps
    acc += swz_xor<1>(acc);
    acc += swz_xor<2>(acc);
    acc += swz_xor<4>(acc);
    acc += swz_xor<8>(acc);
    acc += swz_xor<16>(acc);
    if (lane == 0) atomicAdd(out, acc);
}

extern "C" void kernel_launch(void* const* d_in, const int* in_sizes, int n_in,
                              void* d_out, int out_size, void* d_ws, size_t ws_size,
                              hipStream_t stream) {
    const float* pred = (const float*)d_in[0];
    const int* mask = (const int*)d_in[1];
    // ws layout: [0,512) cnt | [512,1024) sy | [1024,1536) sx (ints), then float4 table[512]
    int* wsInt = (int*)d_ws;
    int* cnt = wsInt;
    int* sy = wsInt + BN * L;
    int* sx = wsInt + 2 * BN * L;
    float4* table = (float4*)((char*)d_ws + 3 * BN * L * sizeof(int));
    float* out = (float*)d_out;

    centroid_init<<<6, 256, 0, stream>>>(wsInt, out);
    centroid_accum<<<BN * 64, 256, 0, stream>>>(mask, cnt, sy, sx);
    centroid_table<<<2, 256, 0, stream>>>(cnt, sy, sx, table);
    centroid_main<<<BN * 32, 256, 0, stream>>>(pred, table, out);
}